// GATConv3_22454089024232
// MI455X (gfx1250) — compile-verified
//
#include <hip/hip_runtime.h>

#define N_NODES 2708
#define IN_CH   1433
#define OUT_CH  128

typedef __attribute__((ext_vector_type(2))) float v2f;
typedef __attribute__((ext_vector_type(8))) float v8f;

#define NEG_INF (-__builtin_inff())

// ---------------------------------------------------------------------------
// Shared WMMA micro-tile: C += A16x16 * B16x16 as 4 x (16x16x4) f32 WMMA.
// A-frag layout: lanes 0-15 -> K pair (0,1), lanes 16-31 -> K pair (2,3), M = lane&15.
// B-frag layout: mirrored; C/D: VGPR r holds M=r (lanes 0-15) / M=r+8 (lanes 16-31).
// ---------------------------------------------------------------------------
__device__ __forceinline__ v8f wmma_tile16(const float (*Amat)[17],
                                           const float (*Bmat)[128],
                                           int l15, int lh, int colbase, v8f c) {
#pragma unroll
    for (int kk = 0; kk < 16; kk += 4) {
        v2f a, b;
        a.x = Amat[l15][kk + 2 * lh];
        a.y = Amat[l15][kk + 2 * lh + 1];
        b.x = Bmat[kk + 2 * lh][colbase];
        b.y = Bmat[kk + 2 * lh + 1][colbase];
        c = __builtin_amdgcn_wmma_f32_16x16x4_f32(false, a, false, b,
                                                  (short)0, c, false, false);
    }
    return c;
}

// ---------------------------------------------------------------------------
// Kernel 1: Kmat[N,128] = X[N,1433] @ W[128,1433]^T   (WMMA f32 16x16x4)
// block = 256 threads = 8 waves; block b -> rows 16b.., wave w -> cols 16w..
// Tail k-tile peeled: hot loop has zero bounds checks.
// ---------------------------------------------------------------------------
__global__ void k_gemm_xw(const float* __restrict__ X, const float* __restrict__ W,
                          float* __restrict__ Kmat) {
    __shared__ float Xs[16][17];
    __shared__ float Ws[16][128];
    const int t    = threadIdx.x;
    const int lane = t & 31;
    const int wv   = t >> 5;
    const int l15  = lane & 15;
    const int lh   = lane >> 4;
    const int row0 = blockIdx.x * 16;

    // per-thread staging coords (hoisted out of the loop)
    const int r    = t >> 4;
    const int kk   = t & 15;
    int xrow = row0 + r;
    if (xrow >= N_NODES) xrow = N_NODES - 1;    // clamp: garbage rows never stored
    const float* xp  = X + (size_t)xrow * IN_CH + kk;
    const int wrow   = t >> 1;
    const int kk8    = (t & 1) * 8;
    const float* wp0 = W + (size_t)wrow * IN_CH + kk8;

    v8f c = {0.f, 0.f, 0.f, 0.f, 0.f, 0.f, 0.f, 0.f};

    const int kFull = IN_CH / 16;               // 89 full tiles
    for (int kt = 0; kt < kFull; ++kt) {
        const int k0 = kt * 16;
        Xs[r][kk] = xp[k0];                     // guard-free
#pragma unroll
        for (int q = 0; q < 8; ++q) Ws[kk8 + q][wrow] = wp0[k0 + q];
        __syncthreads();
        c = wmma_tile16(Xs, Ws, l15, lh, wv * 16 + l15, c);
        __syncthreads();
    }
    // tail tile (k0 = 1424, 9 valid k)
    {
        const int k0 = kFull * 16;
        Xs[r][kk] = (k0 + kk < IN_CH) ? xp[k0] : 0.f;
#pragma unroll
        for (int q = 0; q < 8; ++q)
            Ws[kk8 + q][wrow] = (k0 + kk8 + q < IN_CH) ? wp0[k0 + q] : 0.f;
        __syncthreads();
        c = wmma_tile16(Xs, Ws, l15, lh, wv * 16 + l15, c);
    }
#pragma unroll
    for (int rr = 0; rr < 8; ++rr) {
        int row = row0 + rr + 8 * lh;
        if (row < N_NODES) Kmat[(size_t)row * OUT_CH + wv * 16 + l15] = c[rr];
    }
}

// ---------------------------------------------------------------------------
// Kernel 2: k1 = K @ a1^T, k2 = K @ a2^T
// ---------------------------------------------------------------------------
__global__ void k_attn_vec(const float* __restrict__ Kmat, const float* __restrict__ a1,
                           const float* __restrict__ a2, float* __restrict__ k1,
                           float* __restrict__ k2) {
    int row = blockIdx.x * blockDim.x + threadIdx.x;
    if (row >= N_NODES) return;
    const float4* kr = (const float4*)(Kmat + (size_t)row * OUT_CH);
    const float4* A1 = (const float4*)a1;
    const float4* A2 = (const float4*)a2;
    float s1 = 0.f, s2 = 0.f;
#pragma unroll
    for (int i = 0; i < OUT_CH / 4; ++i) {
        float4 kv = kr[i], x1 = A1[i], x2 = A2[i];
        s1 += kv.x * x1.x + kv.y * x1.y + kv.z * x1.z + kv.w * x1.w;
        s2 += kv.x * x2.x + kv.y * x2.y + kv.z * x2.z + kv.w * x2.w;
    }
    k1[row] = s1;
    k2[row] = s2;
}

// ---------------------------------------------------------------------------
// Kernel 3: u1[p] = sum_i U[i,p]*k1[i] ; s[p] = sum_i U[i,p]   (chunked atomics)
// ---------------------------------------------------------------------------
__global__ void k_ut_reduce(const float* __restrict__ U, const float* __restrict__ k1,
                            float* __restrict__ u1, float* __restrict__ sv) {
    int p = blockIdx.x * 256 + threadIdx.x;
    if (p >= N_NODES) return;
    const int chunk = (N_NODES + gridDim.y - 1) / gridDim.y;
    int i0 = blockIdx.y * chunk;
    int i1 = i0 + chunk;
    if (i1 > N_NODES) i1 = N_NODES;
    float acc1 = 0.f, acc2 = 0.f;
    for (int i = i0; i < i1; ++i) {
        float u = U[(size_t)i * N_NODES + p];   // coalesced across p
        acc1 += u * k1[i];
        acc2 += u;
    }
    atomicAdd(&u1[p], acc1);
    atomicAdd(&sv[p], acc2);
}

// ---------------------------------------------------------------------------
// Kernel 4: u1 *= lmbd ; sv *= lmbd
// ---------------------------------------------------------------------------
__global__ void k_scale(const float* __restrict__ lmbd, float* __restrict__ u1,
                        float* __restrict__ sv) {
    int p = blockIdx.x * 256 + threadIdx.x;
    if (p < N_NODES) {
        float l = lmbd[p];
        u1[p] *= l;
        sv[p] *= l;
    }
}

// ---------------------------------------------------------------------------
// Kernel 5: v1[i] = sum_p U[i,p]*w1[p] ; v2[i] = sum_p U[i,p]*w2[p]
// ---------------------------------------------------------------------------
__global__ void k_u_gemv(const float* __restrict__ U, const float* __restrict__ w1,
                         const float* __restrict__ w2, float* __restrict__ v1,
                         float* __restrict__ v2) {
    __shared__ float red1[256];
    __shared__ float red2[256];
    const int row = blockIdx.x;
    const float* Ur = U + (size_t)row * N_NODES;
    float a = 0.f, b = 0.f;
    for (int p = threadIdx.x; p < N_NODES; p += 256) {
        float u = Ur[p];
        a += u * w1[p];
        b += u * w2[p];
    }
    red1[threadIdx.x] = a;
    red2[threadIdx.x] = b;
    __syncthreads();
    for (int s = 128; s > 0; s >>= 1) {
        if (threadIdx.x < s) {
            red1[threadIdx.x] += red1[threadIdx.x + s];
            red2[threadIdx.x] += red2[threadIdx.x + s];
        }
        __syncthreads();
    }
    if (threadIdx.x == 0) {
        v1[row] = red1[0];
        v2[row] = red2[0];
    }
}

// ---------------------------------------------------------------------------
// Kernel 6: fused masked softmax (rank-2 E2) + H = alpha @ K  (WMMA)
// E2[i,j] = |v1[i] + v2[i]*k2[j]|  (abs >= 0 -> leaky-ReLU is identity)
// Tail j-tile peeled; K tile staged with float4 (Kmat rows are 512B-aligned).
// ---------------------------------------------------------------------------
__global__ void k_softmax_gemm(const int* __restrict__ A, const float* __restrict__ k2,
                               const float* __restrict__ v1, const float* __restrict__ v2,
                               const float* __restrict__ Kmat, float* __restrict__ H) {
    __shared__ float As[16][17];
    __shared__ float Ks[16][128];
    __shared__ float Ms[16][16], Ss[16][16];
    __shared__ float rowM[16], rowInvS[16], rowV1[16], rowV2[16];

    const int t    = threadIdx.x;
    const int row0 = blockIdx.x * 16;

    if (t < 16) {
        int row = row0 + t;
        rowV1[t] = (row < N_NODES) ? v1[row] : 0.f;
        rowV2[t] = (row < N_NODES) ? v2[row] : 0.f;
    }
    __syncthreads();

    // ---- phase 1: per-row online masked-softmax stats (16 lanes per row) ----
    {
        int r = t >> 4, cidx = t & 15;
        int row = row0 + r;
        float m = NEG_INF, s = 0.f;
        if (row < N_NODES) {
            const float a = rowV1[r], b = rowV2[r];
            const int* Ar = A + (size_t)row * N_NODES;
            for (int j = cidx; j < N_NODES; j += 16) {
                if (Ar[j] != 0) {
                    float e  = fabsf(a + b * k2[j]);
                    float mn = fmaxf(m, e);
                    s = s * __expf(m - mn) + __expf(e - mn);
                    m = mn;
                }
            }
        }
        Ms[r][cidx] = m;
        Ss[r][cidx] = s;
    }
    __syncthreads();
    if (t < 16) {
        float M = NEG_INF, S = 0.f;
#pragma unroll
        for (int c2 = 0; c2 < 16; ++c2) {
            float mc = Ms[t][c2], sc = Ss[t][c2];
            float mn = fmaxf(M, mc);
            if (mn > NEG_INF) {   // avoid (-inf) - (-inf) = NaN
                S = S * __expf(M - mn) + sc * __expf(mc - mn);
                M = mn;
            }
        }
        rowM[t]    = M;
        rowInvS[t] = (S > 0.f) ? 1.f / S : 0.f;
    }
    __syncthreads();

    // ---- phase 2: H = alpha @ K via WMMA f32 16x16x4 ----
    const int lane = t & 31;
    const int wv   = t >> 5;
    const int l15  = lane & 15;
    const int lh   = lane >> 4;

    // hoisted staging coords
    const int ra = t >> 4, jja = t & 15;            // alpha-tile element
    int arow = row0 + ra;
    if (arow >= N_NODES) arow = N_NODES - 1;        // clamp: garbage rows never stored
    const int* Abase = A + (size_t)arow * N_NODES;
    const int jjk = t >> 4;                         // K-tile row
    const int col = (t & 15) * 8;                   // K-tile col base (float4-aligned)

    v8f c = {0.f, 0.f, 0.f, 0.f, 0.f, 0.f, 0.f, 0.f};

    const int jFull = N_NODES / 16;                 // 169 full tiles
    for (int jt = 0; jt < jFull; ++jt) {
        const int j0 = jt * 16;
        // alpha tile (guard-free)
        {
            int j = j0 + jja;
            float al = 0.f;
            if (Abase[j] != 0) {
                float e = fabsf(rowV1[ra] + rowV2[ra] * k2[j]);
                al = __expf(e - rowM[ra]) * rowInvS[ra];
            }
            As[ra][jja] = al;
        }
        // K tile: two b128 loads + b128 LDS stores
        {
            const float4* kr = (const float4*)(Kmat + (size_t)(j0 + jjk) * OUT_CH + col);
            float4 p0 = kr[0], p1 = kr[1];
            *(float4*)&Ks[jjk][col]     = p0;
            *(float4*)&Ks[jjk][col + 4] = p1;
        }
        __syncthreads();
        c = wmma_tile16(As, Ks, l15, lh, wv * 16 + l15, c);
        __syncthreads();
    }
    // tail tile (j0 = 2704, 4 valid j)
    {
        const int j0 = jFull * 16;
        int j = j0 + jja;
        float al = 0.f;
        if (j < N_NODES && Abase[j] != 0) {
            float e = fabsf(rowV1[ra] + rowV2[ra] * k2[j]);
            al = __expf(e - rowM[ra]) * rowInvS[ra];
        }
        As[ra][jja] = al;
        float4 p0 = {0.f, 0.f, 0.f, 0.f}, p1 = {0.f, 0.f, 0.f, 0.f};
        if (j0 + jjk < N_NODES) {
            const float4* kr = (const float4*)(Kmat + (size_t)(j0 + jjk) * OUT_CH + col);
            p0 = kr[0];
            p1 = kr[1];
        }
        *(float4*)&Ks[jjk][col]     = p0;
        *(float4*)&Ks[jjk][col + 4] = p1;
        __syncthreads();
        c = wmma_tile16(As, Ks, l15, lh, wv * 16 + l15, c);
    }
#pragma unroll
    for (int rr = 0; rr < 8; ++rr) {
        int row = row0 + rr + 8 * lh;
        if (row < N_NODES) H[(size_t)row * OUT_CH + wv * 16 + l15] = c[rr];
    }
}

// ---------------------------------------------------------------------------
extern "C" void kernel_launch(void* const* d_in, const int* in_sizes, int n_in,
                              void* d_out, int out_size, void* d_ws, size_t ws_size,
                              hipStream_t stream) {
    const float* X    = (const float*)d_in[0];
    const int*   A    = (const int*)  d_in[1];
    const float* U    = (const float*)d_in[2];
    const float* W    = (const float*)d_in[3];
    const float* a1   = (const float*)d_in[4];
    const float* a2   = (const float*)d_in[5];
    const float* lmbd = (const float*)d_in[6];
    float* H = (float*)d_out;

    // workspace layout (floats): Kmat | k1 | k2 | u1 | sv | v1 | v2
    float* ws   = (float*)d_ws;
    float* Kmat = ws;                                // offset 0 -> 16B-aligned rows
    float* k1   = Kmat + (size_t)N_NODES * OUT_CH;
    float* k2   = k1 + N_NODES;
    float* u1   = k2 + N_NODES;
    float* sv   = u1 + N_NODES;
    float* v1   = sv + N_NODES;
    float* v2   = v1 + N_NODES;

    const int rowTiles = (N_NODES + 15) / 16;        // 170
    const int vecBlks  = (N_NODES + 255) / 256;      // 11

    k_gemm_xw<<<rowTiles, 256, 0, stream>>>(X, W, Kmat);
    k_attn_vec<<<vecBlks, 256, 0, stream>>>(Kmat, a1, a2, k1, k2);
    hipMemsetAsync(u1, 0, 2 * (size_t)N_NODES * sizeof(float), stream);
    dim3 g3(vecBlks, 16);
    k_ut_reduce<<<g3, 256, 0, stream>>>(U, k1, u1, sv);
    k_scale<<<vecBlks, 256, 0, stream>>>(lmbd, u1, sv);
    k_u_gemv<<<N_NODES, 256, 0, stream>>>(U, u1, sv, v1, v2);
    k_softmax_gemm<<<rowTiles, 256, 0, stream>>>(A, k2, v1, v2, Kmat, H);
}